// custom_loss_91182155694238
// MI455X (gfx1250) — compile-verified
//
#include <hip/hip_runtime.h>

#define GH 13
#define GW 13
#define NBOX 5
#define NELEM 85
#define NCLS 80
#define NT 50
#define APB (GH * GW * NBOX)            // 845 anchors per batch image
#define TPB 256                          // 8 waves, one anchor per lane
#define NBLKX ((APB + TPB - 1) / TPB)    // 4 blocks per batch image
#define BATCH 256

__constant__ float c_anchors[10] = {
    0.57273f, 0.677385f, 1.87446f, 2.06253f, 3.33843f,
    5.47434f, 7.88282f, 3.52778f, 9.77052f, 9.16828f};

__device__ __forceinline__ float sigmoidf(float x) {
  return 1.0f / (1.0f + __expf(-x));
}

typedef int v4i __attribute__((ext_vector_type(4)));

__global__ __launch_bounds__(TPB) void yolo_loss_partial(
    const float* __restrict__ y_true, const float* __restrict__ y_pred,
    const float* __restrict__ true_boxes, float* __restrict__ acc) {
  __shared__ __align__(16) float s_tb[NT * 4];
  __shared__ float s_acc[8];

  const int lane = threadIdx.x & 31;
  const int wave = threadIdx.x >> 5;
  const int b = blockIdx.y;

  if (threadIdx.x < 8) s_acc[threadIdx.x] = 0.0f;

  // Stage this batch image's 50 true boxes (800 B) into LDS, asynchronously.
  const float* tb_src = true_boxes + (size_t)b * (NT * 4);
  if (wave == 0) {
#if __has_builtin(__builtin_amdgcn_global_load_async_to_lds_b128)
    __builtin_amdgcn_global_load_async_to_lds_b128(
        (v4i*)(tb_src + lane * 4), (v4i*)(&s_tb[lane * 4]), 0, 0);
    if (lane < NT - 32)
      __builtin_amdgcn_global_load_async_to_lds_b128(
          (v4i*)(tb_src + (lane + 32) * 4), (v4i*)(&s_tb[(lane + 32) * 4]), 0,
          0);
#if __has_builtin(__builtin_amdgcn_s_wait_asynccnt)
    __builtin_amdgcn_s_wait_asynccnt(0);
#else
    asm volatile("s_wait_asynccnt 0" ::: "memory");
#endif
#else
    for (int i = lane; i < NT; i += 32) {
      float4 v = ((const float4*)tb_src)[i];
      ((float4*)s_tb)[i] = v;
    }
#endif
  }
  __syncthreads();

  float a_lxy = 0.f, a_lwh = 0.f, a_lcf = 0.f, a_lcl = 0.f;
  float a_nbc = 0.f, a_nbf = 0.f, a_nbl = 0.f;

  const int aib = blockIdx.x * TPB + threadIdx.x; // this lane's anchor
  if (aib < APB) {
    const int an = aib % NBOX;
    const int cell = aib / NBOX;
    const int gx = cell % GW;
    const int gy = cell / GW;
    const size_t base = ((size_t)b * APB + aib) * NELEM;
    const float* ytp = y_true + base;
    const float* ypp = y_pred + base;

    // Header: box + conf raw values.
    const float tx = __builtin_nontemporal_load(ytp + 0);
    const float ty = __builtin_nontemporal_load(ytp + 1);
    const float tw = __builtin_nontemporal_load(ytp + 2);
    const float th = __builtin_nontemporal_load(ytp + 3);
    const float tc = __builtin_nontemporal_load(ytp + 4);
    const float rx = ypp[0], ry = ypp[1], rw = ypp[2], rh = ypp[3],
                rc = ypp[4];

    const float px = sigmoidf(rx) + (float)gx;
    const float py = sigmoidf(ry) + (float)gy;
    const float pw = __expf(rw) * c_anchors[2 * an + 0];
    const float ph = __expf(rh) * c_anchors[2 * an + 1];
    const float pc = sigmoidf(rc);
    const float phw = pw * 0.5f, phh = ph * 0.5f;
    const float parea = pw * ph;

    // IoU vs the anchor's own true box.
    float iw = fminf(px + phw, tx + tw * 0.5f) - fmaxf(px - phw, tx - tw * 0.5f);
    float ih = fminf(py + phh, ty + th * 0.5f) - fmaxf(py - phh, ty - th * 0.5f);
    iw = fmaxf(iw, 0.f); ih = fmaxf(ih, 0.f);
    const float inter = iw * ih;
    const float iou = inter / (parea + tw * th - inter);
    const float tbc = iou * tc; // true_box_conf

    // best_iou < 0.6  <=>  for every box: inter < 0.6 * union (union > 0).
    // LDS broadcast reads: every lane reads the same box -> conflict-free.
    bool noobj = true;
#pragma unroll 5
    for (int i = 0; i < NT; ++i) {
      const float4 bb = ((const float4*)s_tb)[i];
      float iw2 = fminf(px + phw, bb.x + bb.z * 0.5f) -
                  fmaxf(px - phw, bb.x - bb.z * 0.5f);
      float ih2 = fminf(py + phh, bb.y + bb.w * 0.5f) -
                  fmaxf(py - phh, bb.y - bb.w * 0.5f);
      iw2 = fmaxf(iw2, 0.f); ih2 = fmaxf(ih2, 0.f);
      const float in2 = iw2 * ih2;
      const float un2 = parea + bb.z * bb.w - in2;
      noobj = noobj && (in2 < 0.6f * un2);
    }
    const float conf_mask = (noobj ? 1.f : 0.f) * (1.f - tc) + tbc * 5.0f;

    // Pass 1 over classes: max of y_pred logits; y_true argmax carrying the
    // matching y_pred logit (strict > == first-max-wins like jnp.argmax).
    float m = -__builtin_inff();
    float bv = -__builtin_inff();
    float bp = 0.f;
#pragma unroll 8
    for (int c = 0; c < NCLS; ++c) {
      const float pv = ypp[5 + c];
      const float tv = __builtin_nontemporal_load(ytp + 5 + c);
      m = fmaxf(m, pv);
      if (tv > bv) { bv = tv; bp = pv; }
    }
    // Pass 2: sum of exp; y_pred lines are still in near caches (RT hint).
    float ssum = 0.f;
#pragma unroll 8
    for (int c = 0; c < NCLS; ++c) ssum += __expf(ypp[5 + c] - m);
    const float ce = (m + __logf(ssum)) - bp;

    a_lxy = ((tx - px) * (tx - px) + (ty - py) * (ty - py)) * tc;
    a_lwh = ((tw - pw) * (tw - pw) + (th - ph) * (th - ph)) * tc;
    a_lcf = (tbc - pc) * (tbc - pc) * conf_mask;
    a_lcl = ce * tc;
    a_nbc = (tc > 0.f) ? 1.f : 0.f;
    a_nbf = (conf_mask > 0.f) ? 1.f : 0.f;
    a_nbl = a_nbc;
  }

  // Wave32 shuffle-tree reduction of the 7 partials, then LDS, then global.
  for (int off = 16; off > 0; off >>= 1) {
    a_lxy += __shfl_xor(a_lxy, off, 32);
    a_lwh += __shfl_xor(a_lwh, off, 32);
    a_lcf += __shfl_xor(a_lcf, off, 32);
    a_lcl += __shfl_xor(a_lcl, off, 32);
    a_nbc += __shfl_xor(a_nbc, off, 32);
    a_nbf += __shfl_xor(a_nbf, off, 32);
    a_nbl += __shfl_xor(a_nbl, off, 32);
  }
  if (lane == 0) {
    atomicAdd(&s_acc[0], a_lxy);
    atomicAdd(&s_acc[1], a_lwh);
    atomicAdd(&s_acc[2], a_lcf);
    atomicAdd(&s_acc[3], a_lcl);
    atomicAdd(&s_acc[4], a_nbc);
    atomicAdd(&s_acc[5], a_nbf);
    atomicAdd(&s_acc[6], a_nbl);
  }
  __syncthreads();
  if (threadIdx.x < 7) atomicAdd(&acc[threadIdx.x], s_acc[threadIdx.x]);
}

__global__ void zero_acc_kernel(float* acc) {
  if (threadIdx.x < 8) acc[threadIdx.x] = 0.f;
}

__global__ void finalize_kernel(const float* __restrict__ acc,
                                float* __restrict__ out) {
  if (threadIdx.x == 0) {
    const float nbc = acc[4] + 1e-6f;
    const float nbf = acc[5] + 1e-6f;
    const float nbl = acc[6] + 1e-6f;
    out[0] = acc[0] / nbc * 0.5f + acc[1] / nbc * 0.5f +
             acc[2] / nbf * 0.5f + acc[3] / nbl;
  }
}

extern "C" void kernel_launch(void* const* d_in, const int* in_sizes, int n_in,
                              void* d_out, int out_size, void* d_ws,
                              size_t ws_size, hipStream_t stream) {
  const float* y_true = (const float*)d_in[0];
  const float* y_pred = (const float*)d_in[1];
  const float* true_boxes = (const float*)d_in[2];
  float* out = (float*)d_out;
  float* acc = (float*)d_ws;

  zero_acc_kernel<<<1, 32, 0, stream>>>(acc);
  dim3 grid(NBLKX, BATCH, 1);
  yolo_loss_partial<<<grid, TPB, 0, stream>>>(y_true, y_pred, true_boxes, acc);
  finalize_kernel<<<1, 32, 0, stream>>>(acc, out);
}